// PairLinear_MixerBlock_39264591020239
// MI455X (gfx1250) — compile-verified
//
#include <hip/hip_runtime.h>

// PairLinear_MixerBlock on MI455X (gfx1250, wave32).
//
// 10-stage butterfly factored into three block-diagonal passes:
//   Pass A: stages 0-3 -> dense 16x16 blocks on contiguous 16-col groups (64 mats)
//   Pass B: stages 4-7 -> dense 16x16 blocks on stride-16 col groups     (64 mats)
//   Pass C: stages 8-9 -> 4x4 blocks on stride-256 col groups            (256 mats)
// A/B run on V_WMMA_F32_16X16X4_F32 (full fp32). Group matrices are stored in
// WMMA B-fragment order -> two coalesced b128 loads per group per lane.
// The 16x1024 tile is moved by the Tensor Data Mover (TENSOR_LOAD_TO_LDS) with
// D# padding of 1 dword per 256 dwords, which realizes the bank-conflict-free
// row stride of 1028 floats in hardware: LDS dword(r,c) = r*1028 + c + (c>>8).
// Traffic floor = 2 * 64 MiB / 23.3 TB/s ~= 5.8 us; compute stays under it.

typedef __attribute__((ext_vector_type(2))) float v2f;
typedef __attribute__((ext_vector_type(8))) float v8f;
typedef __attribute__((ext_vector_type(4))) unsigned int u32x4;
typedef __attribute__((ext_vector_type(8))) int i32x8;
typedef __attribute__((ext_vector_type(4))) int i32x4;

#define DIM      1024
#define NROWS    16
#define TSTRIDE  1028   // padded row stride in floats (TDM inserts 1 dw / 256 dw)
#define LDS_BYTES (NROWS * TSTRIDE * 4)
#define BLOCK    128    // 4 waves cooperating on one 16-row tile

__device__ __forceinline__ int tcol(int c) { return c + (c >> 8); }

// ---------------------------------------------------------------------------
// Precompute composed group matrices from Ws (10,512,2,2).
// Stage i's 2x2 matrix index for the pair based at element e (bit i clear) is
// p = e with bit i removed. Right-multiply accumulation per matrix row:
//   G[r][u],G[r][v] <- G[r][u]*W00 + G[r][v]*W10 , G[r][u]*W01 + G[r][v]*W11
// MAf/MBf are written in WMMA B-fragment order:
//   frag[g][lane][q], lane = h*16+n, q = 2*kk + bit, holding G[K][n],
//   K = kk*4 + 2*h + bit.   MC stays row-major [lo8][t][tp].
// ---------------------------------------------------------------------------
__global__ void pl_build_mats(const float* __restrict__ Ws,
                              float* __restrict__ MAf,
                              float* __restrict__ MBf,
                              float* __restrict__ MC) {
  int t = blockIdx.x * blockDim.x + threadIdx.x;
  if (t < 2048) {                      // MAf / MBf: 64 groups x 16 K-rows each
    bool isB = t >= 1024;
    int tt = t & 1023;
    int g = tt >> 4, r = tt & 15;      // r = K row of the 16x16 group matrix
    float vec[16];
    #pragma unroll
    for (int j = 0; j < 16; ++j) vec[j] = (j == r) ? 1.0f : 0.0f;
    if (!isB) {                        // stages 0-3, element(k) = g*16 + k
      for (int i = 0; i < 4; ++i) {
        int ls = 1 << i;
        for (int u = 0; u < 16; ++u) {
          if (u & ls) continue;
          int v  = u + ls;
          int eu = g * 16 + u;
          int p  = ((eu >> (i + 1)) << i) | (eu & (ls - 1));
          const float* wm = Ws + (size_t)(i * 512 + p) * 4;
          float a = vec[u], b = vec[v];
          vec[u] = a * wm[0] + b * wm[2];
          vec[v] = a * wm[1] + b * wm[3];
        }
      }
    } else {                           // stages 4-7, element(k) = hi*256+k*16+lo
      int hi = g >> 4, lo = g & 15;
      for (int i = 4; i < 8; ++i) {
        int lb = 1 << (i - 4);
        int gs = 1 << i;
        for (int u = 0; u < 16; ++u) {
          if (u & lb) continue;
          int v  = u + lb;
          int eu = hi * 256 + u * 16 + lo;
          int p  = ((eu >> (i + 1)) << i) | (eu & (gs - 1));
          const float* wm = Ws + (size_t)(i * 512 + p) * 4;
          float a = vec[u], b = vec[v];
          vec[u] = a * wm[0] + b * wm[2];
          vec[v] = a * wm[1] + b * wm[3];
        }
      }
    }
    int kk  = r >> 2;
    int rem = r & 3;
    int h   = rem >> 1;
    int bit = rem & 1;
    int q   = 2 * kk + bit;
    float* dst = (isB ? MBf : MAf) + g * 256 + h * 16 * 8 + q;
    #pragma unroll
    for (int nn = 0; nn < 16; ++nn) dst[nn * 8] = vec[nn];
  } else if (t < 3072) {               // MC: 256 groups x 4 rows, stages 8-9
    int tt  = t - 2048;
    int lo8 = tt >> 2, r = tt & 3;
    float vec[4];
    #pragma unroll
    for (int j = 0; j < 4; ++j) vec[j] = (j == r) ? 1.0f : 0.0f;
    for (int i = 8; i < 10; ++i) {
      int lb = 1 << (i - 8);
      int gs = 1 << i;
      for (int u = 0; u < 4; ++u) {
        if (u & lb) continue;
        int v  = u + lb;
        int eu = lo8 + 256 * u;
        int p  = ((eu >> (i + 1)) << i) | (eu & (gs - 1));
        const float* wm = Ws + (size_t)(i * 512 + p) * 4;
        float a = vec[u], b = vec[v];
        vec[u] = a * wm[0] + b * wm[2];
        vec[v] = a * wm[1] + b * wm[3];
      }
    }
    #pragma unroll
    for (int j = 0; j < 4; ++j) MC[(lo8 * 4 + r) * 4 + j] = vec[j];
  }
}

// ---------------------------------------------------------------------------
// Main kernel: 4 waves per block cooperate on one 16-row tile (groups split
// 16/wave per pass; disjoint columns, barrier between passes).
// Fragment layouts (ISA 7.12.2, wave32): 2-VGPR A (16x4) VGPR j holds
// K = j + 2*(lane>>4); C/D VGPR r holds M = r + 8*(lane>>4), N = lane&15.
// ---------------------------------------------------------------------------
__global__ void pl_mixer_wmma(const float* __restrict__ x,
                              float* __restrict__ out,
                              const float* __restrict__ MAf,
                              const float* __restrict__ MBf,
                              const float* __restrict__ MC) {
  extern __shared__ float tile[];            // padded: dword(r,c) = r*1028+c+(c>>8)
  const int tid     = threadIdx.x;
  const int lane    = tid & 31;
  const int w       = tid >> 5;              // wave id 0..3
  const int rowBase = blockIdx.x * NROWS;
  const int r = lane & 15;                   // A-fragment row (M)
  const int h = lane >> 4;                   // wave half: K offset = 2h
  const int n = lane & 15;                   // B/D column (N)

  // LDS byte offset of the dynamic-shared tile.
  unsigned ldsBase = (unsigned)(unsigned long long)
      (__attribute__((address_space(3))) float*)tile;

#if __has_builtin(__builtin_amdgcn_tensor_load_to_lds)
  // ---- TDM: one descriptor moves the whole 16x1024 tile, HW-padded ----
  if (w == 0) {
    unsigned long long ga = (unsigned long long)(x + (size_t)rowBase * DIM);
    u32x4 g0;
    g0[0] = 1u;                                        // count=1 (valid user D#)
    g0[1] = ldsBase;                                   // lds_addr (bytes)
    g0[2] = (unsigned)(ga & 0xFFFFFFFFull);            // global_addr[31:0]
    g0[3] = (unsigned)((ga >> 32) & 0x1FFFFFFull)      // global_addr[56:32]
          | (2u << 30);                                // type = 2 ("image")
    i32x8 g1;
    g1[0] = (2 << 16)                                  // data_size = 4 B
          | (1 << 20)                                  // pad_enable
          | (7 << 22)                                  // pad_interval: 256 dwords
          | (0 << 25);                                 // pad_amount: 1 dword
    g1[1] = (int)(1024u << 16);                        // tensor_dim0 = 1024
    g1[2] = (int)(16u << 16);                          // tensor_dim1 = 16
    g1[3] = (int)(1024u << 16);                        // tile_dim0 = 1024
    g1[4] = 16;                                        // tile_dim1 = 16, tile_dim2 = 0
    g1[5] = 1024;                                      // tensor_dim0_stride = 1024
    g1[6] = 0;
    g1[7] = 0;
    i32x4 gz4 = {0, 0, 0, 0};                          // 2-D tensor: groups 2/3 unused
    i32x8 gz8 = {0, 0, 0, 0, 0, 0, 0, 0};
    __builtin_amdgcn_tensor_load_to_lds(g0, g1, gz4, gz4, gz8, 0);
    __builtin_amdgcn_s_wait_tensorcnt(0);
  }
#else
  // ---- fallback: coalesced b128 global loads + scalar LDS stores ----
  for (int rr = 0; rr < NROWS; ++rr) {
    const float* src = x + (size_t)(rowBase + rr) * DIM;
    for (int s = 0; s < DIM / (BLOCK * 4); ++s) {
      int col = (s * BLOCK + tid) * 4;
      float4 v = *(const float4*)(src + col);
      int base = rr * TSTRIDE + tcol(col);
      tile[base + 0] = v.x; tile[base + 1] = v.y;
      tile[base + 2] = v.z; tile[base + 3] = v.w;
    }
  }
#endif
  __syncthreads();

  // ---- Pass A: stages 0-3, dense 16x16 on contiguous column groups ----
  for (int g = w * 16; g < w * 16 + 16; ++g) {
    const float* Mg = MAf + g * 256 + lane * 8;
    float4 f0 = *(const float4*)(Mg);        // b-frags kk=0,1
    float4 f1 = *(const float4*)(Mg + 4);    // b-frags kk=2,3
    v2f b0 = {f0.x, f0.y}, b1 = {f0.z, f0.w};
    v2f b2 = {f1.x, f1.y}, b3 = {f1.z, f1.w};
    int cbp = g * 16 + (g >> 4);             // padded column base (16 | 256 => no split)
    const float* arow = &tile[r * TSTRIDE + cbp + 2 * h];
    v2f a0 = *(const v2f*)(arow + 0);
    v2f a1 = *(const v2f*)(arow + 4);
    v2f a2 = *(const v2f*)(arow + 8);
    v2f a3 = *(const v2f*)(arow + 12);
    v8f acc = {};
    acc = __builtin_amdgcn_wmma_f32_16x16x4_f32(false, a0, false, b0, (short)0, acc, false, false);
    acc = __builtin_amdgcn_wmma_f32_16x16x4_f32(false, a1, false, b1, (short)0, acc, false, false);
    acc = __builtin_amdgcn_wmma_f32_16x16x4_f32(false, a2, false, b2, (short)0, acc, false, false);
    acc = __builtin_amdgcn_wmma_f32_16x16x4_f32(false, a3, false, b3, (short)0, acc, false, false);
    #pragma unroll
    for (int rr = 0; rr < 8; ++rr)
      tile[(rr + 8 * h) * TSTRIDE + cbp + n] = acc[rr];
  }
  __syncthreads();

  // ---- Pass B: stages 4-7, dense 16x16 on stride-16 column groups ----
  for (int gb = w * 16; gb < w * 16 + 16; ++gb) {
    int hi = gb >> 4, lo = gb & 15;
    int cbp = hi * 256 + hi + lo;            // tcol(hi*256+lo); k*16+lo < 256 in-block
    const float* Mg = MBf + gb * 256 + lane * 8;
    float4 f0 = *(const float4*)(Mg);
    float4 f1 = *(const float4*)(Mg + 4);
    v2f b0 = {f0.x, f0.y}, b1 = {f0.z, f0.w};
    v2f b2 = {f1.x, f1.y}, b3 = {f1.z, f1.w};
    const float* arow = &tile[r * TSTRIDE + cbp + 2 * h * 16];
    v2f a0, a1, a2, a3;
    a0.x = arow[0 * 16];  a0.y = arow[1 * 16];
    a1.x = arow[4 * 16];  a1.y = arow[5 * 16];
    a2.x = arow[8 * 16];  a2.y = arow[9 * 16];
    a3.x = arow[12 * 16]; a3.y = arow[13 * 16];
    v8f acc = {};
    acc = __builtin_amdgcn_wmma_f32_16x16x4_f32(false, a0, false, b0, (short)0, acc, false, false);
    acc = __builtin_amdgcn_wmma_f32_16x16x4_f32(false, a1, false, b1, (short)0, acc, false, false);
    acc = __builtin_amdgcn_wmma_f32_16x16x4_f32(false, a2, false, b2, (short)0, acc, false, false);
    acc = __builtin_amdgcn_wmma_f32_16x16x4_f32(false, a3, false, b3, (short)0, acc, false, false);
    #pragma unroll
    for (int rr = 0; rr < 8; ++rr)
      tile[(rr + 8 * h) * TSTRIDE + cbp + n * 16] = acc[rr];
  }
  __syncthreads();

  // ---- Pass C: stages 8-9, 4x4 mix at stride 256; coalesced global store ----
  for (int it = 0; it < (NROWS * 256) / BLOCK; ++it) {
    int idx  = it * BLOCK + tid;
    int rr   = idx >> 8;
    int lo8  = idx & 255;
    const float* Mc = MC + lo8 * 16;
    const float* trow = &tile[rr * TSTRIDE + lo8];   // padded: +257 per t-step
    float v0 = trow[0 * 257];
    float v1 = trow[1 * 257];
    float v2 = trow[2 * 257];
    float v3 = trow[3 * 257];
    float* dst = out + (size_t)(rowBase + rr) * DIM + lo8;
    #pragma unroll
    for (int tp = 0; tp < 4; ++tp) {
      dst[tp * 256] = v0 * Mc[0 * 4 + tp] + v1 * Mc[1 * 4 + tp] +
                      v2 * Mc[2 * 4 + tp] + v3 * Mc[3 * 4 + tp];
    }
  }
}

extern "C" void kernel_launch(void* const* d_in, const int* in_sizes, int n_in,
                              void* d_out, int out_size, void* d_ws, size_t ws_size,
                              hipStream_t stream) {
  (void)in_sizes; (void)n_in; (void)out_size; (void)ws_size;
  const float* x  = (const float*)d_in[0];   // (16384, 1024) f32
  const float* Ws = (const float*)d_in[1];   // (10, 512, 2, 2) f32
  float* out = (float*)d_out;

  float* wsf = (float*)d_ws;
  float* MAf = wsf;                 // 64 * 256 floats, fragment order
  float* MBf = wsf + 64 * 256;      // 64 * 256 floats, fragment order
  float* MC  = wsf + 2 * 64 * 256;  // 256 * 16 floats

  pl_build_mats<<<12, 256, 0, stream>>>(Ws, MAf, MBf, MC);
  pl_mixer_wmma<<<16384 / NROWS, BLOCK, LDS_BYTES, stream>>>(x, out, MAf, MBf, MC);
}